// ManifoldHyperConnectionLayer_74517682586067
// MI455X (gfx1250) — compile-verified
//
#include <hip/hip_runtime.h>
#include <hip/hip_bf16.h>

typedef __attribute__((ext_vector_type(2))) float v2f;
typedef __attribute__((ext_vector_type(8))) float v8f;
typedef __attribute__((ext_vector_type(4))) int   v4i;

#define N_STREAMS 4
#define HIDDEN 1024
#define DMODEL 4096
#define SINKHORN_ITERS 20
#define TAU 0.05f
#define EPS_SUM 1e-6f
#define RMS_EPS 1e-8f

// ---- CDNA5 async global->LDS support detection --------------------------
#if defined(__has_builtin)
#if __has_builtin(__builtin_amdgcn_global_load_async_to_lds_b128)
#define USE_ASYNC_LDS 1
#endif
#if __has_builtin(__builtin_amdgcn_s_wait_asynccnt)
#define HAVE_WAIT_ASYNC_BUILTIN 1
#endif
#endif
#ifndef USE_ASYNC_LDS
#define USE_ASYNC_LDS 0
#endif
#ifndef HAVE_WAIT_ASYNC_BUILTIN
#define HAVE_WAIT_ASYNC_BUILTIN 0
#endif

typedef __attribute__((address_space(1))) v4i* gptr_v4i;
typedef __attribute__((address_space(3))) v4i* lptr_v4i;

__device__ __forceinline__ void wait_async_zero() {
#if USE_ASYNC_LDS
#if HAVE_WAIT_ASYNC_BUILTIN
    __builtin_amdgcn_s_wait_asynccnt(0);
#else
    asm volatile("s_wait_asynccnt 0x0" ::: "memory");
#endif
#endif
}

__device__ __forceinline__ void stage_b128(const float* __restrict__ gsrc,
                                           float* __restrict__ ldst) {
#if USE_ASYNC_LDS
    __builtin_amdgcn_global_load_async_to_lds_b128(
        (gptr_v4i)gsrc, (lptr_v4i)ldst, 0 /*imm offset*/, 0 /*cpol*/);
#else
    const float4 v = *(const float4*)gsrc;
    ldst[0] = v.x; ldst[1] = v.y; ldst[2] = v.z; ldst[3] = v.w;
#endif
}

__device__ __forceinline__ float wave_sum(float v) {
#pragma unroll
    for (int m = 16; m > 0; m >>= 1) v += __shfl_xor(v, m, 32);
    return v;
}

__device__ __forceinline__ float sigmoidf(float z) {
    return 1.0f / (1.0f + __expf(-z));
}

// -------------------------------------------------------------------------
// Kernel 1: per-token gating. One wave32 per token.
//   - RMS factor + 24 length-4096 dot products (lane-strided, shfl reduce)
//   - 4x4 log-Sinkhorn (20 iters) redundantly in registers on every lane
//   - writes gates[t*24]: [0..3]=H_pre_w [4..7]=H_post_w [8..23]=H_res(o,i)
//   - writes x_layer[t, c] = sum_i x[t,i,c] * H_pre_w[i]
// -------------------------------------------------------------------------
__global__ __launch_bounds__(256) void gating_kernel(
    const float* __restrict__ x, const float* __restrict__ rw,
    const float* __restrict__ Wpre, const float* __restrict__ Wpost,
    const float* __restrict__ Wres,
    const float* __restrict__ bpre, const float* __restrict__ bpost,
    const float* __restrict__ bres,
    const float* __restrict__ apre, const float* __restrict__ apost,
    const float* __restrict__ ares,
    float* __restrict__ gates, float* __restrict__ xlayer, int T)
{
    const int wid  = threadIdx.x >> 5;
    const int lane = threadIdx.x & 31;
    const int t = blockIdx.x * 8 + wid;
    if (t >= T) return;

    const float* xb = x + (size_t)t * DMODEL;

    float s2 = 0.0f;
    float dp[4] = {0.f,0.f,0.f,0.f};
    float dq[4] = {0.f,0.f,0.f,0.f};
    float dr[16];
#pragma unroll
    for (int k = 0; k < 16; ++k) dr[k] = 0.0f;

    for (int e = lane; e < DMODEL; e += 32) {
        float xv = xb[e];
        s2 += xv * xv;
        float xw = xv * rw[e];
#pragma unroll
        for (int k = 0; k < 4; ++k)  dp[k] += xw * Wpre [k * DMODEL + e];
#pragma unroll
        for (int k = 0; k < 4; ++k)  dq[k] += xw * Wpost[k * DMODEL + e];
#pragma unroll
        for (int k = 0; k < 16; ++k) dr[k] += xw * Wres [k * DMODEL + e];
    }

    s2 = wave_sum(s2);
#pragma unroll
    for (int k = 0; k < 4; ++k)  dp[k] = wave_sum(dp[k]);
#pragma unroll
    for (int k = 0; k < 4; ++k)  dq[k] = wave_sum(dq[k]);
#pragma unroll
    for (int k = 0; k < 16; ++k) dr[k] = wave_sum(dr[k]);

    const float rinv = rsqrtf(s2 * (1.0f / DMODEL) + RMS_EPS);
    const float ap = apre[0], ao = apost[0], ar = ares[0];

    float hp[4], hq[4];
    float sp = 0.f, sq = 0.f;
#pragma unroll
    for (int k = 0; k < 4; ++k) {
        hp[k] = sigmoidf(ap * rinv * dp[k] + bpre[k]);
        hq[k] = 2.0f * sigmoidf(ao * rinv * dq[k] + bpost[k]);
        sp += hp[k]; sq += hq[k];
    }
    const float ipn = 1.0f / (sp + EPS_SUM);
    const float iqn = 1.0f / (sq + EPS_SUM);

    // log-domain Sinkhorn on the 4x4
    float Z[16];
#pragma unroll
    for (int k = 0; k < 16; ++k)
        Z[k] = (ar * rinv * dr[k] + bres[k]) * (1.0f / TAU);

    float u[4] = {0.f,0.f,0.f,0.f}, v[4] = {0.f,0.f,0.f,0.f};
    for (int it = 0; it < SINKHORN_ITERS; ++it) {
#pragma unroll
        for (int o = 0; o < 4; ++o) {       // u = -LSE over i of Z[o,i]+v[i]
            float m = Z[o*4+0] + v[0];
#pragma unroll
            for (int i = 1; i < 4; ++i) m = fmaxf(m, Z[o*4+i] + v[i]);
            float s = 0.f;
#pragma unroll
            for (int i = 0; i < 4; ++i) s += __expf(Z[o*4+i] + v[i] - m);
            u[o] = -(m + __logf(s));
        }
#pragma unroll
        for (int i = 0; i < 4; ++i) {       // v = -LSE over o of Z[o,i]+u[o]
            float m = Z[0*4+i] + u[0];
#pragma unroll
            for (int o = 1; o < 4; ++o) m = fmaxf(m, Z[o*4+i] + u[o]);
            float s = 0.f;
#pragma unroll
            for (int o = 0; o < 4; ++o) s += __expf(Z[o*4+i] + u[o] - m);
            v[i] = -(m + __logf(s));
        }
    }

    if (lane == 0) {
        float* g = gates + (size_t)t * 24;
#pragma unroll
        for (int k = 0; k < 4; ++k) g[k]     = hp[k] * ipn;
#pragma unroll
        for (int k = 0; k < 4; ++k) g[4 + k] = hq[k] * iqn;
#pragma unroll
        for (int o = 0; o < 4; ++o)
#pragma unroll
            for (int i = 0; i < 4; ++i)
                g[8 + o*4 + i] = __expf(Z[o*4+i] + u[o] + v[i]);
    }

    // pre-aggregate streams with H_pre_w (all lanes hold hp/ipn)
    const float w0 = hp[0]*ipn, w1 = hp[1]*ipn, w2 = hp[2]*ipn, w3 = hp[3]*ipn;
    float* xl = xlayer + (size_t)t * HIDDEN;
    for (int c = lane; c < HIDDEN; c += 32) {
        xl[c] = xb[c] * w0 + xb[HIDDEN + c] * w1
              + xb[2*HIDDEN + c] * w2 + xb[3*HIDDEN + c] * w3;
    }
}

// -------------------------------------------------------------------------
// Kernel 2: y[T,1024] = x_layer[T,1024] @ W_sub[1024,1024]^T via f32 WMMA.
// Block: 256 thr = 8 waves in 4(M) x 2(N); block tile 64x32, K-tile 32.
// Double-buffered LDS, tiles staged with GLOBAL_LOAD_ASYNC_TO_LDS_B128
// (ASYNCcnt), 8 x v_wmma_f32_16x16x4_f32 per wave per K-tile.
// -------------------------------------------------------------------------
#define TM 64
#define TN 32
#define TK 32
#define LPAD 4   // pad in floats; keeps every b128 LDS target 16B-aligned

__global__ __launch_bounds__(256) void gemm_kernel(
    const float* __restrict__ A,   // x_layer [T, 1024]
    const float* __restrict__ B,   // W_sub   [1024, 1024], y = A * B^T
    float* __restrict__ Y, int T)
{
    __shared__ float As[2][TM][TK + LPAD];
    __shared__ float Bs[2][TN][TK + LPAD];

    const int tid  = threadIdx.x;
    const int lane = tid & 31;
    const int wid  = tid >> 5;
    const int wm   = wid & 3;       // 0..3  -> M offset 16*wm
    const int wn   = wid >> 2;      // 0..1  -> N offset 16*wn
    const int m0   = blockIdx.x * TM;
    const int n0   = blockIdx.y * TN;
    const int lo   = lane & 15;     // row/col within fragment
    const int hi   = lane >> 4;     // K-pair selector

    // per-thread staging coordinates (3 x b128 per thread per K-tile)
    const int ar0 = tid >> 3,          ac0 = (tid & 7) * 4;          // A row 0..31
    const int ar1 = (tid + 256) >> 3,  ac1 = (tid & 7) * 4;          // A row 32..63
    const int br  = tid >> 3;                                        // B row 0..31 (ar0==br)

    auto issue_tile = [&](int buf, int k0) {
        stage_b128(&A[(size_t)(m0 + ar0) * HIDDEN + k0 + ac0], &As[buf][ar0][ac0]);
        stage_b128(&A[(size_t)(m0 + ar1) * HIDDEN + k0 + ac1], &As[buf][ar1][ac1]);
        stage_b128(&B[(size_t)(n0 + br ) * HIDDEN + k0 + ac0], &Bs[buf][br ][ac0]);
        if (k0 + TK < HIDDEN) {
            __builtin_prefetch(&A[(size_t)(m0 + ar0) * HIDDEN + k0 + TK], 0, 0);
            __builtin_prefetch(&B[(size_t)(n0 + br ) * HIDDEN + k0 + TK], 0, 0);
        }
    };

    v8f acc = {};

    issue_tile(0, 0);                       // prologue: tile 0 in flight

    const int NK = HIDDEN / TK;             // 32 K-tiles
    for (int kt = 0; kt < NK; ++kt) {
        const int cur = kt & 1;
        wait_async_zero();                  // my async writes to buf[cur] done
        __syncthreads();                    // everyone's writes visible; prev
                                            // compute on buf[1-cur] finished
        if (kt + 1 < NK)
            issue_tile(1 - cur, (kt + 1) * TK);  // overlap next fetch w/ WMMA

#pragma unroll
        for (int kk = 0; kk < TK / 4; ++kk) {
            const int k = kk * 4 + 2 * hi;
            v2f a, b;
            a.x = As[cur][wm * 16 + lo][k];
            a.y = As[cur][wm * 16 + lo][k + 1];
            b.x = Bs[cur][wn * 16 + lo][k];
            b.y = Bs[cur][wn * 16 + lo][k + 1];
            acc = __builtin_amdgcn_wmma_f32_16x16x4_f32(
                false, a, false, b, (short)0, acc, false, false);
        }
    }

    // D layout: VGPR r -> M = r + 8*hi, N = lo
    const int col = n0 + wn * 16 + lo;
    const int rbase = m0 + wm * 16 + 8 * hi;
#pragma unroll
    for (int r = 0; r < 8; ++r) {
        Y[(size_t)(rbase + r) * HIDDEN + col] = acc[r];
    }
}

// -------------------------------------------------------------------------
// Kernel 3: out[t,o,c] = sum_i x[t,i,c]*Hres[o,i] + Hpost_w[o]*y[t,c]
// -------------------------------------------------------------------------
__global__ __launch_bounds__(256) void combine_kernel(
    const float* __restrict__ x, const float* __restrict__ y,
    const float* __restrict__ gates, float* __restrict__ out, int total)
{
    int idx = blockIdx.x * blockDim.x + threadIdx.x;
    if (idx >= total) return;
    const int t = idx >> 10;
    const int c = idx & (HIDDEN - 1);
    const float* g = gates + (size_t)t * 24;
    const float yv = y[idx];
    const float* xb = x + (size_t)t * DMODEL + c;
    const float x0 = xb[0], x1 = xb[HIDDEN], x2 = xb[2 * HIDDEN], x3 = xb[3 * HIDDEN];
    float* ob = out + (size_t)t * DMODEL + c;
#pragma unroll
    for (int o = 0; o < 4; ++o) {
        float r = g[4 + o] * yv
                + g[8 + o * 4 + 0] * x0 + g[8 + o * 4 + 1] * x1
                + g[8 + o * 4 + 2] * x2 + g[8 + o * 4 + 3] * x3;
        ob[o * HIDDEN] = r;
    }
}

extern "C" void kernel_launch(void* const* d_in, const int* in_sizes, int n_in,
                              void* d_out, int out_size, void* d_ws, size_t ws_size,
                              hipStream_t stream) {
    const float* x     = (const float*)d_in[0];
    const float* rmsw  = (const float*)d_in[1];
    const float* Wpre  = (const float*)d_in[2];
    const float* Wpost = (const float*)d_in[3];
    const float* Wres  = (const float*)d_in[4];
    const float* bpre  = (const float*)d_in[5];
    const float* bpost = (const float*)d_in[6];
    const float* bres  = (const float*)d_in[7];
    const float* apre  = (const float*)d_in[8];
    const float* apost = (const float*)d_in[9];
    const float* ares  = (const float*)d_in[10];
    const float* Wsub  = (const float*)d_in[11];
    float* out = (float*)d_out;

    const int T = in_sizes[0] / DMODEL;   // 8192

    float* gates  = (float*)d_ws;                       // T*24
    float* xlayer = gates + (size_t)T * 24;             // T*1024
    float* y      = xlayer + (size_t)T * HIDDEN;        // T*1024

    gating_kernel<<<(T + 7) / 8, 256, 0, stream>>>(
        x, rmsw, Wpre, Wpost, Wres, bpre, bpost, bres,
        apre, apost, ares, gates, xlayer, T);

    dim3 ggrid(T / TM, HIDDEN / TN);
    gemm_kernel<<<ggrid, 256, 0, stream>>>(xlayer, Wsub, y, T);

    const int total = T * HIDDEN;
    combine_kernel<<<(total + 255) / 256, 256, 0, stream>>>(x, y, gates, out, total);
}